// VerseAwareEmbeddings_5987184411235
// MI455X (gfx1250) — compile-verified
//
#include <hip/hip_runtime.h>
#include <math.h>
#include <stdint.h>

// ---------------------------------------------------------------------------
// VerseAwareEmbeddings for MI455X (gfx1250, wave32)
//
// out[b,s,:] = LN( emb[token]*32 + PE(s)*w[b,s] + book[type] ) * gamma + beta
//
// Bandwidth-bound (~67 MB total traffic -> ~2.9 us @ 23.3 TB/s). No matmul,
// so no WMMA. CDNA5 features used:
//   * global_load_async_to_lds_b128 + s_wait_asynccnt: stage gamma/beta in
//     LDS, overlapped with phase-1 compute (ASYNCcnt-tracked)
//   * wave32 shuffle scan (verse weights) + butterfly reduction (LN stats)
//   * 128-bit coalesced global loads; NT-hint (th:TH_STORE_NT) output stores
//     so the 33.5 MB write-once stream doesn't evict emb rows from L2
//   * scalar (SMEM) loads for wave-uniform token/book/weight values
// ---------------------------------------------------------------------------

#define VOCAB   32000
#define DIM     1024
#define SEQ     2048
#define BATCH   4
#define LN_EPS  1e-5f

#define WAVES_PER_BLOCK 8
#define TPB (WAVES_PER_BLOCK * 32)

// -2*ln(10000)/1024 : pair-index k -> div_term = exp(k * C2)
#define C2 (-0.0179889851405457f)
#define SQRT_D 32.0f

typedef float v4f __attribute__((ext_vector_type(4)));

// ---------------------------------------------------------------------------
// Kernel 1: verse weights. One wave32 per batch row.
// w = 1.2 where (v != 0) && (v != last nonzero verse strictly before s), else 1.0.
// "last nonzero" is associative -> Hillis-Steele wave scan + cross-chunk carry.
// ---------------------------------------------------------------------------
__global__ __launch_bounds__(32) void verse_weight_kernel(
    const int* __restrict__ verse, float* __restrict__ w)
{
    const int b    = blockIdx.x;
    const int lane = threadIdx.x;          // 0..31 (wave32)
    const int base = b * SEQ;

    int carry = -1;                        // last nonzero verse before this chunk
    for (int c = 0; c < SEQ / 32; ++c) {
        const int s = c * 32 + lane;
        const int v = verse[base + s];
        int incl = (v != 0) ? v : -1;

        // inclusive scan with op(l, r) = (r != -1) ? r : l
        #pragma unroll
        for (int off = 1; off < 32; off <<= 1) {
            const int t = __shfl_up(incl, off, 32);
            if (lane >= off && incl == -1) incl = t;
        }

        const int below = __shfl_up(incl, 1, 32);   // inclusive at lane-1
        const int prev  = (lane == 0) ? carry : ((below != -1) ? below : carry);

        w[base + s] = (v != 0 && v != prev) ? 1.2f : 1.0f;

        const int tail = __shfl(incl, 31, 32);
        carry = (tail != -1) ? tail : carry;
    }
}

// ---------------------------------------------------------------------------
// Kernel 2: fused embed + PE + book + LayerNorm. One token per wave32,
// 8 waves per block. Each lane owns 8 float4 (d = c*128 + lane*4).
// ---------------------------------------------------------------------------
__global__ __launch_bounds__(TPB) void embed_ln_kernel(
    const int*   __restrict__ token_ids,
    const int*   __restrict__ book_types,
    const float* __restrict__ emb_table,
    const float* __restrict__ book_table,
    const float* __restrict__ ln_gamma,
    const float* __restrict__ ln_beta,
    const float* __restrict__ vw_buf,
    float*       __restrict__ out)
{
    __shared__ __align__(16) float s_gamma[DIM];
    __shared__ __align__(16) float s_beta[DIM];

    // ---- Issue async gamma/beta -> LDS (overlaps with phase 1) ------------
    {
        const int      i     = threadIdx.x * 4;                 // element idx
        const uint32_t lds_g = (uint32_t)(uintptr_t)(&s_gamma[i]);
        const uint32_t lds_b = (uint32_t)(uintptr_t)(&s_beta[i]);
        const uint32_t boff  = (uint32_t)(i * sizeof(float));   // byte offset
        asm volatile("global_load_async_to_lds_b128 %0, %1, %2"
                     :: "v"(lds_g), "v"(boff), "s"(ln_gamma) : "memory");
        asm volatile("global_load_async_to_lds_b128 %0, %1, %2"
                     :: "v"(lds_b), "v"(boff), "s"(ln_beta)  : "memory");
    }

    // wave index is uniform within the wave: pin it to an SGPR so the
    // token/book/weight loads can take the scalar (SMEM) path.
    const int wave  = __builtin_amdgcn_readfirstlane(threadIdx.x >> 5);
    const int lane  = threadIdx.x & 31;
    const int token = blockIdx.x * WAVES_PER_BLOCK + wave;      // 0..B*S-1
    const int s     = token & (SEQ - 1);

    const int   tok  = token_ids[token];
    const int   book = book_types[token];
    const float vw   = vw_buf[token];
    const float pos  = (float)s;

    const float* __restrict__ emb_row  = emb_table  + (size_t)tok  * DIM;
    const float* __restrict__ book_row = book_table + (size_t)book * DIM;

    // ---- Phase 1: gather + PE + book, accumulate sum / sumsq --------------
    v4f   xv[8];
    float sum = 0.f, sumsq = 0.f;

    #pragma unroll
    for (int c = 0; c < 8; ++c) {
        const int d = c * 128 + lane * 4;
        const v4f e  = *(const v4f*)(emb_row  + d);
        const v4f bt = *(const v4f*)(book_row + d);

        const float k  = (float)(d >> 1);
        const float a0 = pos * __expf(k * C2);
        const float a1 = pos * __expf((k + 1.0f) * C2);

        v4f x;
        x.x = fmaf(e.x, SQRT_D, fmaf(__sinf(a0), vw, bt.x));
        x.y = fmaf(e.y, SQRT_D, fmaf(__cosf(a0), vw, bt.y));
        x.z = fmaf(e.z, SQRT_D, fmaf(__sinf(a1), vw, bt.z));
        x.w = fmaf(e.w, SQRT_D, fmaf(__cosf(a1), vw, bt.w));
        xv[c] = x;

        sum   += (x.x + x.y) + (x.z + x.w);
        sumsq += fmaf(x.x, x.x, x.y * x.y) + fmaf(x.z, x.z, x.w * x.w);
    }

    // ---- wave32 butterfly reduction ---------------------------------------
    #pragma unroll
    for (int off = 16; off > 0; off >>= 1) {
        sum   += __shfl_xor(sum,   off, 32);
        sumsq += __shfl_xor(sumsq, off, 32);
    }
    const float mu   = sum * (1.0f / (float)DIM);
    const float var  = fmaf(-mu, mu, sumsq * (1.0f / (float)DIM));
    const float rsig = rsqrtf(var + LN_EPS);

    // ---- gamma/beta must be in LDS now ------------------------------------
    asm volatile("s_wait_asynccnt 0x0" ::: "memory");
    __syncthreads();

    // ---- Phase 2: normalize, scale, shift; NT b128 stores -----------------
    float* __restrict__ out_row = out + (size_t)token * DIM;
    #pragma unroll
    for (int c = 0; c < 8; ++c) {
        const int d = c * 128 + lane * 4;
        const v4f g  = *(const v4f*)(&s_gamma[d]);
        const v4f be = *(const v4f*)(&s_beta[d]);
        const v4f x  = xv[c];
        v4f o;
        o.x = fmaf((x.x - mu) * rsig, g.x, be.x);
        o.y = fmaf((x.y - mu) * rsig, g.y, be.y);
        o.z = fmaf((x.z - mu) * rsig, g.z, be.z);
        o.w = fmaf((x.w - mu) * rsig, g.w, be.w);
        // write-once stream: NT hint keeps emb-table rows resident in L2
        __builtin_nontemporal_store(o, (v4f*)(out_row + d));
    }
}

// ---------------------------------------------------------------------------
// Host launcher
// ---------------------------------------------------------------------------
extern "C" void kernel_launch(void* const* d_in, const int* in_sizes, int n_in,
                              void* d_out, int out_size, void* d_ws, size_t ws_size,
                              hipStream_t stream)
{
    const int*   token_ids = (const int*)  d_in[0];
    const int*   verse     = (const int*)  d_in[1];
    const int*   books     = (const int*)  d_in[2];
    const float* emb       = (const float*)d_in[3];
    const float* book_tab  = (const float*)d_in[4];
    const float* gamma     = (const float*)d_in[5];
    const float* beta      = (const float*)d_in[6];
    float*       out       = (float*)d_out;
    float*       wbuf      = (float*)d_ws;   // BATCH*SEQ floats (32 KB)

    verse_weight_kernel<<<BATCH, 32, 0, stream>>>(verse, wbuf);

    const int n_tokens = BATCH * SEQ;
    embed_ln_kernel<<<n_tokens / WAVES_PER_BLOCK, TPB, 0, stream>>>(
        token_ids, books, emb, book_tab, gamma, beta, wbuf, out);
}